// SparseHeightCompressionWithConv_51402168598648
// MI455X (gfx1250) — compile-verified
//
#include <hip/hip_runtime.h>

typedef __attribute__((ext_vector_type(2))) float v2f;
typedef __attribute__((ext_vector_type(8))) float v8f;

#define CIN  64
#define COUT 128
#define BN_EPS 1e-5f

// ---------------------------------------------------------------------------
// Kernel 1: zero the output accumulator (d_out is poisoned by the harness)
// ---------------------------------------------------------------------------
__global__ __launch_bounds__(256) void zero_out_kernel(float4* __restrict__ out,
                                                       int total4) {
  int i = blockIdx.x * 256 + threadIdx.x;
  if (i < total4) out[i] = make_float4(0.f, 0.f, 0.f, 0.f);
}

// ---------------------------------------------------------------------------
// Kernel 2: gather-GEMM via V_WMMA_F32_16X16X4_F32 + atomic scatter-add.
// One wave32 handles 16 input rows x 128 output channels.
// contrib = (A .* mask_tap0) @ W0 + (A .* mask_tap1) @ W1
// ---------------------------------------------------------------------------
__global__ __launch_bounds__(256) void spconv_wmma_scatter(
    const float* __restrict__ feats,     // [N, 64]
    const float* __restrict__ W,         // [2, 64, 128]
    const int*   __restrict__ kz,        // [N]
    const int*   __restrict__ out_ids,   // [N]
    float*       __restrict__ out,       // [N, 128] (zeroed)
    int n_rows) {
  // Stage both weight taps into LDS: 2*64*128 f32 = 64 KB.
  __shared__ float sW[2 * CIN * COUT];
  {
    const float4* Wv  = (const float4*)W;
    float4*       sWv = (float4*)sW;
    for (int i = threadIdx.x; i < (2 * CIN * COUT) / 4; i += 256) sWv[i] = Wv[i];
  }
  __syncthreads();

  const int wave  = threadIdx.x >> 5;
  const int lane  = threadIdx.x & 31;
  const int lrow  = lane & 15;   // row-in-strip (A) / col-in-tile (B,C)
  const int khalf = lane >> 4;   // lanes 16-31 hold the upper K pair
  const long strip = (long)blockIdx.x * 8 + wave;
  const int  base  = (int)(strip * 16);
  if (base >= n_rows) return;

  const int  row    = base + lrow;
  const bool rvalid = row < n_rows;
  const int  rr     = rvalid ? row : base;       // safe address
  const int  tap    = rvalid ? kz[rr] : -1;      // -1 => both masks zero

  // Preload this lane's half of its A row: 16 K-tiles, v2f each.
  // A 16x4 f32 layout: lanes 0-15 -> K={0,1}, lanes 16-31 -> K={2,3}.
  v2f a[16];
  const float* arow = feats + (size_t)rr * CIN + khalf * 2;
#pragma unroll
  for (int kt = 0; kt < 16; ++kt) {
    a[kt].x = arow[kt * 4 + 0];
    a[kt].y = arow[kt * 4 + 1];
  }

  v8f acc[8] = {};  // 8 N-tiles of 16 cols => 128 output channels

#pragma unroll
  for (int t = 0; t < 2; ++t) {
    const float m = (tap == t) ? 1.0f : 0.0f;
    const float* sWt = sW + t * (CIN * COUT);
    for (int kt = 0; kt < 16; ++kt) {
      v2f am;
      am.x = a[kt].x * m;
      am.y = a[kt].y * m;
      const int    k0  = kt * 4 + khalf * 2;     // B K-rows mirror A's lane split
      const float* bp0 = sWt + k0 * COUT + lrow;
#pragma unroll
      for (int nt = 0; nt < 8; ++nt) {
        v2f b;
        b.x = bp0[nt * 16];          // K = k0,   N = nt*16 + lrow
        b.y = bp0[COUT + nt * 16];   // K = k0+1
        acc[nt] = __builtin_amdgcn_wmma_f32_16x16x4_f32(
            /*neg_a=*/false, am, /*neg_b=*/false, b,
            /*c_mod=*/(short)0, acc[nt], /*reuse_a=*/false, /*reuse_b=*/false);
      }
    }
  }

  // C/D layout: VGPR j -> M=j (lanes 0-15) or M=j+8 (lanes 16-31); N = lrow + nt*16.
  // Scatter-add into segment rows with fire-and-forget f32 atomics.
#pragma unroll
  for (int j = 0; j < 8; ++j) {
    const int r2 = base + j + khalf * 8;
    if (r2 < n_rows) {
      const int od  = out_ids[r2];
      float*    dst = out + (size_t)od * COUT + lrow;
#pragma unroll
      for (int nt = 0; nt < 8; ++nt) {
        atomicAdd(dst + nt * 16, acc[nt][j]);
      }
    }
  }
}

// ---------------------------------------------------------------------------
// Kernel 3: BatchNorm (eval) + ReLU in place; zero rows >= n_out.
// ---------------------------------------------------------------------------
__global__ __launch_bounds__(256) void bn_relu_kernel(
    float* __restrict__ out,
    const float* __restrict__ gamma, const float* __restrict__ beta,
    const float* __restrict__ rmean, const float* __restrict__ rvar,
    const int* __restrict__ n_out_p, int n_rows) {
  __shared__ float s_scale[COUT];
  __shared__ float s_bias[COUT];
  if (threadIdx.x < COUT) {
    const float sc = gamma[threadIdx.x] * rsqrtf(rvar[threadIdx.x] + BN_EPS);
    s_scale[threadIdx.x] = sc;
    s_bias[threadIdx.x]  = beta[threadIdx.x] - rmean[threadIdx.x] * sc;
  }
  __syncthreads();

  const int n_out  = *n_out_p;
  const int total4 = n_rows * (COUT / 4);
  const int i = blockIdx.x * 256 + threadIdx.x;
  if (i >= total4) return;

  const int row = i >> 5;            // 32 float4 per row
  const int c   = (i & 31) * 4;
  float4* p = (float4*)out + i;
  float4 v = *p;
  if (row < n_out) {
    float r0 = v.x * s_scale[c + 0] + s_bias[c + 0];
    float r1 = v.y * s_scale[c + 1] + s_bias[c + 1];
    float r2 = v.z * s_scale[c + 2] + s_bias[c + 2];
    float r3 = v.w * s_scale[c + 3] + s_bias[c + 3];
    v.x = r0 > 0.f ? r0 : 0.f;
    v.y = r1 > 0.f ? r1 : 0.f;
    v.z = r2 > 0.f ? r2 : 0.f;
    v.w = r3 > 0.f ? r3 : 0.f;
  } else {
    v = make_float4(0.f, 0.f, 0.f, 0.f);
  }
  *p = v;
}

// ---------------------------------------------------------------------------
extern "C" void kernel_launch(void* const* d_in, const int* in_sizes, int n_in,
                              void* d_out, int out_size, void* d_ws, size_t ws_size,
                              hipStream_t stream) {
  const float* feats   = (const float*)d_in[0];
  const float* W       = (const float*)d_in[1];
  const float* gamma   = (const float*)d_in[2];
  const float* beta    = (const float*)d_in[3];
  const float* rmean   = (const float*)d_in[4];
  const float* rvar    = (const float*)d_in[5];
  const int*   kz      = (const int*)d_in[6];
  const int*   out_ids = (const int*)d_in[7];
  const int*   n_out_p = (const int*)d_in[8];
  float*       out     = (float*)d_out;

  const int n_rows = in_sizes[0] / CIN;

  const int total4 = n_rows * (COUT / 4);
  const int zb = (total4 + 255) / 256;
  zero_out_kernel<<<zb, 256, 0, stream>>>((float4*)out, total4);

  const int strips = (n_rows + 15) / 16;
  const int blocks = (strips + 7) / 8;  // 8 waves per 256-thread block
  spconv_wmma_scatter<<<blocks, 256, 0, stream>>>(feats, W, kz, out_ids, out, n_rows);

  bn_relu_kernel<<<zb, 256, 0, stream>>>(out, gamma, beta, rmean, rvar, n_out_p, n_rows);
}